// NodeLevelSAGPooling_74277164417509
// MI455X (gfx1250) — compile-verified
//
#include <hip/hip_runtime.h>
#include <hip/hip_bf16.h>
#include <math.h>

typedef __attribute__((ext_vector_type(2))) float v2f;
typedef __attribute__((ext_vector_type(8))) float v8f;

#define C_DIM 256
#define N_GRAPHS 512

// ---------- helpers: order-preserving float<->uint encoding for atomicMax ----------
__device__ __forceinline__ unsigned fenc(float f) {
    unsigned u = __float_as_uint(f);
    return (u & 0x80000000u) ? ~u : (u | 0x80000000u);
}
__device__ __forceinline__ float fdec(unsigned u) {
    unsigned b = (u & 0x80000000u) ? (u & 0x7FFFFFFFu) : ~u;
    return __uint_as_float(b);
}

// ---------- 1) init node scratch + per-graph accumulators ----------
__global__ void k_init(float* __restrict__ deg, float* __restrict__ agg,
                       unsigned* __restrict__ gmaxu, float* __restrict__ gden, int N) {
    int i = blockIdx.x * blockDim.x + threadIdx.x;
    if (i < N) {
        deg[i] = 1.0f;   // self loop
        agg[i] = 0.0f;
    }
    if (i < N_GRAPHS) {
        gmaxu[i] = 0u;   // encodes "smaller than any real float"
        gden[i]  = 0.0f;
    }
}

// ---------- 2) degree scatter over edges ----------
__global__ void k_deg(const int* __restrict__ ei, float* __restrict__ deg, int E) {
    int e = blockIdx.x * blockDim.x + threadIdx.x;
    if (e >= E) return;
    int dst = ei[E + e];
    atomicAdd(&deg[dst], 1.0f);
}

// ---------- 3) dinv = rsqrt(max(deg,1)) in place ----------
__global__ void k_dinv(float* __restrict__ deg, int N) {
    int i = blockIdx.x * blockDim.x + threadIdx.x;
    if (i >= N) return;
    deg[i] = rsqrtf(fmaxf(deg[i], 1.0f));
}

// ---------- 4) h = x @ W via V_WMMA_F32_16X16X4_F32 (one wave = 16 rows) ----------
__global__ void k_matvec_wmma(const float* __restrict__ x, const float* __restrict__ W,
                              float* __restrict__ h, int N, int nWaves) {
    int gtid = blockIdx.x * blockDim.x + threadIdx.x;
    int wave = gtid >> 5;
    int lane = threadIdx.x & 31;
    if (wave >= nWaves) return;          // wave-uniform exit: EXEC stays all-1s

    int base = wave * 16;
    int m    = lane & 15;                // A row this lane feeds
    int kb   = (lane >> 4) << 1;         // K sub-offset: 0 for lanes 0-15, 2 for 16-31
    int row  = base + m; if (row >= N) row = N - 1;   // clamp (keeps EXEC full)
    const float* xrow = x + (size_t)row * C_DIM;

    v8f acc = {};
    #pragma unroll 4
    for (int k0 = 0; k0 < C_DIM; k0 += 4) {
        v2f a, b;
        a.x = xrow[k0 + kb];
        a.y = xrow[k0 + kb + 1];
        b.x = W[k0 + kb];                // broadcast W across all 16 B columns
        b.y = W[k0 + kb + 1];
        acc = __builtin_amdgcn_wmma_f32_16x16x4_f32(
            /*neg_a=*/false, a, /*neg_b=*/false, b,
            /*c_mod=*/(short)0, acc, /*reuse_a=*/false, /*reuse_b=*/false);
    }

    // D layout: lane L, VGPR r -> M = r + 8*(L/16), N = L%16. Every column equals h.
    int hi = (lane >> 4) * 8;
    #pragma unroll
    for (int r = 0; r < 8; ++r) {
        int outRow = base + hi + r;
        if ((lane & 15) == r && outRow < N) h[outRow] = acc[r];
    }
}

// ---------- 5) edge aggregation: agg[dst] += dinv[src]*dinv[dst]*h[src] ----------
__global__ void k_edge_agg(const int* __restrict__ ei, const float* __restrict__ dinv,
                           const float* __restrict__ h, float* __restrict__ agg, int E) {
    int e = blockIdx.x * blockDim.x + threadIdx.x;
    if (e >= E) return;
    int src = ei[e];
    int dst = ei[E + e];
    atomicAdd(&agg[dst], dinv[src] * dinv[dst] * h[src]);
}

// ---------- 6) scores = tanh(agg + selfloop + b); per-graph running max ----------
__global__ void k_scores(const float* __restrict__ agg, const float* __restrict__ dinv,
                         const float* __restrict__ h, const float* __restrict__ bias,
                         const int* __restrict__ batch, float* __restrict__ sc,
                         unsigned* __restrict__ gmaxu, int N) {
    int i = blockIdx.x * blockDim.x + threadIdx.x;
    if (i >= N) return;
    float s = tanhf(agg[i] + dinv[i] * dinv[i] * h[i] + bias[0]);
    sc[i] = s;
    atomicMax(&gmaxu[batch[i]], fenc(s));
}

// ---------- 7) e = exp(s - max); per-graph denom ----------
__global__ void k_exp(const float* __restrict__ sc, const int* __restrict__ batch,
                      const unsigned* __restrict__ gmaxu, float* __restrict__ ebuf,
                      float* __restrict__ gden, int N) {
    int i = blockIdx.x * blockDim.x + threadIdx.x;
    if (i >= N) return;
    int g = batch[i];
    float e = expf(sc[i] - fdec(gmaxu[g]));
    ebuf[i] = e;
    atomicAdd(&gden[g], e);
}

// ---------- 8) normalized scores -> output ----------
__global__ void k_norm(const float* __restrict__ ebuf, const int* __restrict__ batch,
                       const float* __restrict__ gden, float* __restrict__ out_scores, int N) {
    int i = blockIdx.x * blockDim.x + threadIdx.x;
    if (i >= N) return;
    out_scores[i] = ebuf[i] / gden[batch[i]];
}

// ---------- 9) pooled[g] = sum over graph segment of w_i * x[i]; 1 block/graph ----------
__global__ void k_pool(const float* __restrict__ x, const int* __restrict__ batch,
                       const float* __restrict__ w, float* __restrict__ pooled, int N) {
    int g = blockIdx.x;      // 0..511
    int c = threadIdx.x;     // 0..255 : channel
    // lower_bound(batch, g) and lower_bound(batch, g+1) in the sorted batch array
    int lo = 0, hi = N;
    while (lo < hi) { int mid = (lo + hi) >> 1; if (batch[mid] < g) lo = mid + 1; else hi = mid; }
    int start = lo;
    lo = start; hi = N;
    while (lo < hi) { int mid = (lo + hi) >> 1; if (batch[mid] < g + 1) lo = mid + 1; else hi = mid; }
    int end = lo;
    float acc = 0.0f;
    for (int i = start; i < end; ++i)
        acc += w[i] * x[(size_t)i * C_DIM + c];   // coalesced across the 256 threads
    pooled[(size_t)g * C_DIM + c] = acc;
}

extern "C" void kernel_launch(void* const* d_in, const int* in_sizes, int n_in,
                              void* d_out, int out_size, void* d_ws, size_t ws_size,
                              hipStream_t stream) {
    (void)n_in; (void)out_size; (void)ws_size;
    const float* x     = (const float*)d_in[0];
    const int*   ei    = (const int*)d_in[1];    // [2,E] flat: row0=src, row1=dst
    const int*   batch = (const int*)d_in[2];
    const float* W     = (const float*)d_in[3];
    const float* bias  = (const float*)d_in[4];

    const int N = in_sizes[0] / C_DIM;   // 100000
    const int E = in_sizes[1] / 2;       // 3200000

    float* pooled     = (float*)d_out;                   // [G, C]
    float* out_scores = pooled + (size_t)N_GRAPHS * C_DIM; // [N]

    float*    ws    = (float*)d_ws;
    float*    dinv  = ws;               // N  (holds deg first, then rsqrt)
    float*    h     = ws + N;           // N
    float*    agg   = ws + 2 * (size_t)N; // N
    float*    sc    = ws + 3 * (size_t)N; // N
    float*    ebuf  = ws + 4 * (size_t)N; // N
    unsigned* gmaxu = (unsigned*)(ws + 5 * (size_t)N);    // G
    float*    gden  = ws + 5 * (size_t)N + N_GRAPHS;      // G

    const int TB = 256;
    int nBlkN = (N + TB - 1) / TB;
    int nBlkE = (E + TB - 1) / TB;
    int nWaves = (N + 15) / 16;
    int nThrMV = nWaves * 32;
    int nBlkMV = (nThrMV + TB - 1) / TB;

    k_init<<<nBlkN, TB, 0, stream>>>(dinv, agg, gmaxu, gden, N);
    k_deg<<<nBlkE, TB, 0, stream>>>(ei, dinv, E);
    k_dinv<<<nBlkN, TB, 0, stream>>>(dinv, N);
    k_matvec_wmma<<<nBlkMV, TB, 0, stream>>>(x, W, h, N, nWaves);
    k_edge_agg<<<nBlkE, TB, 0, stream>>>(ei, dinv, h, agg, E);
    k_scores<<<nBlkN, TB, 0, stream>>>(agg, dinv, h, bias, batch, sc, gmaxu, N);
    k_exp<<<nBlkN, TB, 0, stream>>>(sc, batch, gmaxu, ebuf, gden, N);
    k_norm<<<nBlkN, TB, 0, stream>>>(ebuf, batch, gden, out_scores, N);
    k_pool<<<N_GRAPHS, C_DIM, 0, stream>>>(x, batch, out_scores, pooled, N);
}